// JointDecoder_81398220193861
// MI455X (gfx1250) — compile-verified
//
#include <hip/hip_runtime.h>
#include <hip/hip_bf16.h>
#include <math.h>

// ---------------------------------------------------------------------------
// JointDecoder fused masked-slot-attention decoder for gfx1250 (MI455X).
// f32 end-to-end. All four GEMM-like stages run on V_WMMA_F32_16X16X4_F32:
//   q  = x @ Wq.T + bq            (64-deep)
//   lg = q @ k.T * SCALE          (64-deep, 9 slots zero-padded to 16)
//   H  = softmax(lg) @ slots_full (16-deep, padded)
//   xo = H @ out_w.T + bo         (64-deep)
// Only the 9-wide mask/softmax/sigma runs on VALU.
// LDS row strides chosen so every WMMA operand load is bank-conflict-free
// (stride 68 == 4 mod 64 -> 16 rows x 4 consecutive floats cover 64 banks).
// ---------------------------------------------------------------------------

typedef float v2f __attribute__((ext_vector_type(2)));
typedef float v8f __attribute__((ext_vector_type(8)));

#define N_PTS   262144      // points per batch
#define BATCH   2
#define DDIM    64          // D == S == 64
#define KSLOT   8
#define K1      9           // K+1 (empty slot prepended)
#define TILE_M  128         // points per workgroup
#define SCALE_Q 0.125f      // 64^-0.5

#define STR     68          // row stride (floats) for sBuf / sW  (68 % 64 == 4)
#define SSTR    68          // row stride for padded slots_full
#define LSTR    18          // row stride for logits/weights tile

__device__ __forceinline__ v8f wmma_f32_16x16x4(v2f a, v2f b, v8f c) {
    // (neg_a, A, neg_b, B, c_mod, C, reuse_a, reuse_b)
    return __builtin_amdgcn_wmma_f32_16x16x4_f32(false, a, false, b,
                                                 (short)0, c, false, false);
}

// ---------------------------------------------------------------------------
// Setup (tiny, once per batch): slots_full = [empty; slots] and the
// pre-scaled, zero-padded, transposed key matrix kT[d][slot] so the big
// kernel's logits WMMA needs no masking and no extra scale pass.
// ---------------------------------------------------------------------------
__global__ void jd_setup_kernel(const float* __restrict__ slots,      // [B,8,64]
                                const float* __restrict__ empty_slot, // [64]
                                const float* __restrict__ Wk_w,       // [64,64] (d,s)
                                const float* __restrict__ Wk_b,       // [64]
                                float* __restrict__ ws_slotsF,        // [B,9,64]
                                float* __restrict__ ws_kmatT)         // [B,64,16]
{
    const int b = blockIdx.x;
    const int t = threadIdx.x;

    for (int i = t; i < K1 * DDIM; i += blockDim.x) {
        const int kk = i >> 6, s = i & 63;
        const float v = (kk == 0) ? empty_slot[s]
                                  : slots[((size_t)b * KSLOT + (kk - 1)) * DDIM + s];
        ws_slotsF[(size_t)b * K1 * DDIM + i] = v;
    }
    for (int i = t; i < DDIM * 16; i += blockDim.x) {
        const int d = i >> 4, n = i & 15;
        float v = 0.f;
        if (n < K1) {
            const float* srow = (n == 0) ? empty_slot
                                         : &slots[((size_t)b * KSLOT + (n - 1)) * DDIM];
            float acc = Wk_b[d];
            #pragma unroll 8
            for (int s = 0; s < DDIM; ++s) acc += srow[s] * Wk_w[d * DDIM + s];
            v = acc * SCALE_Q;
        }
        ws_kmatT[(size_t)b * DDIM * 16 + i] = v;
    }
}

// ---------------------------------------------------------------------------
// Fused main kernel: 256 threads (8 wave32), 128-point tile.
// ---------------------------------------------------------------------------
__global__ __launch_bounds__(256, 1)
void jd_main_kernel(const float* __restrict__ x,        // [B*N,64]
                    const float* __restrict__ coor,     // [B*N,3]
                    const float* __restrict__ Wq_w,     // [64,64] (n,k)
                    const float* __restrict__ Wq_b,     // [64]
                    const float* __restrict__ out_wM,   // [64,64] (n,k)
                    const float* __restrict__ out_b,    // [64]
                    const float* __restrict__ dscale,   // [1]
                    const float* __restrict__ ws_slotsF,// [B,9,64]
                    const float* __restrict__ ws_kmatT, // [B,64,16]
                    float* __restrict__ o_xo,           // [B*N,64]
                    float* __restrict__ o_w,            // [B*N,9]
                    float* __restrict__ o_sigma)        // [B*N]
{
    __shared__ __align__(16) float sBuf[TILE_M * STR];   // x -> Q -> H   (34816 B)
    __shared__ __align__(16) float uWL[DDIM * STR];      // Wq -> sL -> out_w (17408 B)
    __shared__ __align__(16) float sK[DDIM * 16];        // kT padded     ( 4096 B)
    __shared__ __align__(16) float sS[16 * SSTR];        // slots padded  ( 4352 B)
    __shared__ float sBq[DDIM];
    __shared__ float sBo[DDIM];

    float* const sW = uWL;          // weight view (stride STR)
    float* const sL = uWL;          // logits/attn-weight view (stride LSTR)

    const int tilesPerB = N_PTS / TILE_M;
    const int b    = blockIdx.x / tilesPerB;
    const int trow = blockIdx.x % tilesPerB;
    const long g0  = (long)b * N_PTS + (long)trow * TILE_M;

    const int t = threadIdx.x;

    // ---------------- prologue: stage inputs into LDS ----------------
    {
        const float4* xs = (const float4*)(x + g0 * DDIM);
        for (int i = t; i < TILE_M * 16; i += 256) {
            const int r = i >> 4, c = i & 15;
            *(float4*)&sBuf[r * STR + c * 4] = xs[i];
        }
    }
    {
        const float4* wq4 = (const float4*)Wq_w;
        for (int i = t; i < DDIM * 16; i += 256) {
            const int r = i >> 4, c = i & 15;
            *(float4*)&sW[r * STR + c * 4] = wq4[i];
        }
    }
    {
        const float4* kt4 = (const float4*)(ws_kmatT + (size_t)b * DDIM * 16);
        float4* sk4 = (float4*)sK;
        for (int i = t; i < DDIM * 16 / 4; i += 256) sk4[i] = kt4[i];
    }
    {
        const float* sf = ws_slotsF + (size_t)b * K1 * DDIM;
        for (int i = t; i < 16 * SSTR; i += 256) {
            const int r = i / SSTR, c = i % SSTR;
            sS[i] = (r < K1 && c < DDIM) ? sf[r * DDIM + c] : 0.f;
        }
    }
    if (t < DDIM)           sBq[t]      = Wq_b[t];
    else if (t < 2 * DDIM)  sBo[t - 64] = out_b[t - 64];

    if (trow + 1 < tilesPerB && t == 0)                  // global_prefetch_b8
        __builtin_prefetch(x + (g0 + TILE_M) * DDIM, 0, 1);

    __syncthreads();

    const int lane = t & 31;
    const int wave = t >> 5;
    const int m    = lane & 15;    // column / A-row within 16-tile
    const int kh   = lane >> 4;    // which K-pair this lane carries
    const int rowA = wave * 16 + m;

    // ============ stage 1: Q = x @ Wq.T + bq  (64 WMMA/wave) ============
    {
        v8f acc[4];
        #pragma unroll
        for (int nt = 0; nt < 4; ++nt) {
            const float bv = sBq[nt * 16 + m];
            v8f c;
            #pragma unroll
            for (int j = 0; j < 8; ++j) c[j] = bv;
            acc[nt] = c;
        }
        #pragma unroll
        for (int ks = 0; ks < 16; ++ks) {
            const int kb = ks * 4 + kh * 2;
            const v2f a = *(const v2f*)&sBuf[rowA * STR + kb];
            #pragma unroll
            for (int nt = 0; nt < 4; ++nt) {
                const v2f bb = *(const v2f*)&sW[(nt * 16 + m) * STR + kb];
                acc[nt] = wmma_f32_16x16x4(a, bb, acc[nt]);
            }
        }
        #pragma unroll
        for (int nt = 0; nt < 4; ++nt)
            #pragma unroll
            for (int j = 0; j < 8; ++j)
                sBuf[(wave * 16 + kh * 8 + j) * STR + nt * 16 + m] = acc[nt][j];
    }
    __syncthreads();   // Wq dead -> uWL becomes the logits tile

    // ====== stage 2a: logits = Q @ kT (scale folded, 16 WMMA/wave) ======
    {
        v8f accL = 0.f;
        #pragma unroll
        for (int ks = 0; ks < 16; ++ks) {
            const int kb = ks * 4 + kh * 2;
            const v2f a = *(const v2f*)&sBuf[rowA * STR + kb];
            v2f bb;
            bb.x = sK[(kb + 0) * 16 + m];
            bb.y = sK[(kb + 1) * 16 + m];
            accL = wmma_f32_16x16x4(a, bb, accL);
        }
        #pragma unroll
        for (int j = 0; j < 8; ++j)
            sL[(wave * 16 + kh * 8 + j) * LSTR + m] = accL[j];
        // columns 9..15 hold exact zeros (zero-padded kT) -> ready-made
        // zero padding for the H GEMM's A matrix.
    }
    __syncthreads();

    // ====== stage 2b: mask / softmax / sigma (one thread per point) ======
    const float dsc = __expf(dscale[0]);
    if (t < TILE_M) {
        const int  r  = t;
        const long gr = g0 + r;

        float lg[K1];
        #pragma unroll
        for (int kk = 0; kk < K1; ++kk) lg[kk] = sL[r * LSTR + kk];

        const float cx = coor[gr * 3 + 0];
        const float cy = coor[gr * 3 + 1];
        const float cz = coor[gr * 3 + 2];
        const bool oob = (fabsf(cx) > 1.f) || (fabsf(cy) > 1.f) || (fabsf(cz) > 1.f);
        if (oob) {
            #pragma unroll
            for (int kk = 2; kk < K1; ++kk) lg[kk] = -INFINITY;
        }

        float mx = lg[0];
        #pragma unroll
        for (int kk = 1; kk < K1; ++kk) mx = fmaxf(mx, lg[kk]);
        float wgt[K1]; float sum = 0.f;
        #pragma unroll
        for (int kk = 0; kk < K1; ++kk) { const float e = __expf(lg[kk] - mx); wgt[kk] = e; sum += e; }
        const float inv = 1.f / sum;

        float sig = 0.f;
        #pragma unroll
        for (int kk = 1; kk < K1; ++kk) sig += fmaxf(lg[kk], 0.f) * wgt[kk];
        o_sigma[gr] = sig * inv * dsc;

        #pragma unroll
        for (int kk = 0; kk < K1; ++kk) {
            const float wn = wgt[kk] * inv;
            o_w[gr * K1 + kk] = wn;
            sL[r * LSTR + kk] = wn;        // cols 9..15 stay zero
        }
    }
    __syncthreads();

    // ====== stage 2c: H = w @ slots_full  (16 WMMA/wave, K=16 padded) ======
    {
        v8f acc[4];
        #pragma unroll
        for (int nt = 0; nt < 4; ++nt) acc[nt] = 0.f;
        #pragma unroll
        for (int ks = 0; ks < 4; ++ks) {
            const int kb = ks * 4 + kh * 2;
            const v2f a = *(const v2f*)&sL[rowA * LSTR + kb];
            #pragma unroll
            for (int nt = 0; nt < 4; ++nt) {
                v2f bb;
                bb.x = sS[(kb + 0) * SSTR + nt * 16 + m];
                bb.y = sS[(kb + 1) * SSTR + nt * 16 + m];
                acc[nt] = wmma_f32_16x16x4(a, bb, acc[nt]);
            }
        }
        #pragma unroll
        for (int nt = 0; nt < 4; ++nt)
            #pragma unroll
            for (int j = 0; j < 8; ++j)
                sBuf[(wave * 16 + kh * 8 + j) * STR + nt * 16 + m] = acc[nt][j];
    }
    __syncthreads();   // sL dead -> uWL becomes out_w

    {
        const float4* ow4 = (const float4*)out_wM;
        for (int i = t; i < DDIM * 16; i += 256) {
            const int r = i >> 4, c = i & 15;
            *(float4*)&sW[r * STR + c * 4] = ow4[i];
        }
    }
    __syncthreads();

    // ============ stage 3: xo = H @ out_w.T + bo (64 WMMA/wave) ============
    {
        v8f acc[4];
        #pragma unroll
        for (int nt = 0; nt < 4; ++nt) {
            const float bv = sBo[nt * 16 + m];
            v8f c;
            #pragma unroll
            for (int j = 0; j < 8; ++j) c[j] = bv;
            acc[nt] = c;
        }
        #pragma unroll
        for (int ks = 0; ks < 16; ++ks) {
            const int kb = ks * 4 + kh * 2;
            const v2f a = *(const v2f*)&sBuf[rowA * STR + kb];
            #pragma unroll
            for (int nt = 0; nt < 4; ++nt) {
                const v2f bb = *(const v2f*)&sW[(nt * 16 + m) * STR + kb];
                acc[nt] = wmma_f32_16x16x4(a, bb, acc[nt]);
            }
        }
        float* outBase = o_xo + (g0 + wave * 16) * DDIM;
        #pragma unroll
        for (int nt = 0; nt < 4; ++nt)
            #pragma unroll
            for (int j = 0; j < 8; ++j)
                outBase[(kh * 8 + j) * DDIM + nt * 16 + m] = acc[nt][j];
    }
}

// ---------------------------------------------------------------------------
extern "C" void kernel_launch(void* const* d_in, const int* in_sizes, int n_in,
                              void* d_out, int out_size, void* d_ws, size_t ws_size,
                              hipStream_t stream) {
    const float* point_feats = (const float*)d_in[0];
    // d_in[1] = points_emb (unused by the reference block)
    const float* slots   = (const float*)d_in[2];
    const float* coor    = (const float*)d_in[3];
    const float* empty   = (const float*)d_in[4];
    const float* Wq_w    = (const float*)d_in[5];
    const float* Wq_b    = (const float*)d_in[6];
    const float* Wk_w    = (const float*)d_in[7];
    const float* Wk_b    = (const float*)d_in[8];
    const float* out_wM  = (const float*)d_in[9];
    const float* out_b   = (const float*)d_in[10];
    const float* dsc     = (const float*)d_in[11];
    // d_in[12] = Nr (unused)

    float* ws_slotsF = (float*)d_ws;                            // B*9*64
    float* ws_kmatT  = ws_slotsF + (size_t)BATCH * K1 * DDIM;   // B*64*16

    float* o_xo    = (float*)d_out;                             // [B*N,64]
    float* o_w     = o_xo + (size_t)BATCH * N_PTS * DDIM;       // [B*N,9]
    float* o_sigma = o_w  + (size_t)BATCH * N_PTS * K1;         // [B*N]

    jd_setup_kernel<<<BATCH, 256, 0, stream>>>(slots, empty, Wk_w, Wk_b,
                                               ws_slotsF, ws_kmatT);

    jd_main_kernel<<<BATCH * (N_PTS / TILE_M), 256, 0, stream>>>(
        point_feats, coor, Wq_w, Wq_b, out_wM, out_b, dsc,
        ws_slotsF, ws_kmatT, o_xo, o_w, o_sigma);
}